// triplet_loss_45071386804286
// MI455X (gfx1250) — compile-verified
//
#include <hip/hip_runtime.h>

typedef float v4f __attribute__((ext_vector_type(4)));

namespace {
constexpr int   kD             = 512;                     // feature dim
constexpr int   kRows          = 32768;                   // 98304 / 3 triplets
constexpr float kAlpha         = 0.2f;
constexpr int   kWavesPerBlock = 8;                       // 256 threads = 8 wave32
constexpr int   kBlocks        = kRows / kWavesPerBlock;  // 4096
}  // namespace

// One wave32 per triplet row. Each lane streams 16 floats of A/P/N via four
// b128 non-temporal loads (512B coalesced per wave-instruction), accumulates
// squared diffs, reduces across the wave with xor-shuffles, then the block
// writes one deterministic partial (sum of hard losses, hard count).
__global__ __launch_bounds__(256) void triplet_partial_kernel(
    const float* __restrict__ x, float* __restrict__ psum, int* __restrict__ pcnt) {
  __shared__ float s_sum[kWavesPerBlock];
  __shared__ int   s_cnt[kWavesPerBlock];

  const int wave = threadIdx.x >> 5;   // wave32: 32-lane waves on gfx1250
  const int lane = threadIdx.x & 31;
  const int row  = blockIdx.x * kWavesPerBlock + wave;

  const v4f* __restrict__ arow = (const v4f*)(x + (size_t)row * kD);
  const v4f* __restrict__ prow = (const v4f*)(x + (size_t)(row + kRows) * kD);
  const v4f* __restrict__ nrow = (const v4f*)(x + (size_t)(row + 2 * kRows) * kD);

  float pos = 0.0f, neg = 0.0f;
#pragma unroll
  for (int c = 0; c < kD / (4 * 32); ++c) {   // 4 chunks of 32 lanes * float4
    const int i = c * 32 + lane;
    v4f a = __builtin_nontemporal_load(&arow[i]);  // streamed once: TH=NT,
    v4f p = __builtin_nontemporal_load(&prow[i]);  // don't thrash the 192MB L2
    v4f n = __builtin_nontemporal_load(&nrow[i]);
    v4f dp = a - p;
    v4f dn = a - n;
    pos = fmaf(dp.x, dp.x, pos); pos = fmaf(dp.y, dp.y, pos);
    pos = fmaf(dp.z, dp.z, pos); pos = fmaf(dp.w, dp.w, pos);
    neg = fmaf(dn.x, dn.x, neg); neg = fmaf(dn.y, dn.y, neg);
    neg = fmaf(dn.z, dn.z, neg); neg = fmaf(dn.w, dn.w, neg);
  }

  // wave32 butterfly reduction (explicit width 32 — NOT 64 on CDNA5)
#pragma unroll
  for (int off = 16; off > 0; off >>= 1) {
    pos += __shfl_xor(pos, off, 32);
    neg += __shfl_xor(neg, off, 32);
  }

  if (lane == 0) {
    const float pd   = sqrtf(pos);
    const float nd   = sqrtf(neg);
    const bool  hard = (nd - pd) < kAlpha;
    s_sum[wave] = hard ? (pd - nd + kAlpha) : 0.0f;
    s_cnt[wave] = hard ? 1 : 0;
  }
  __syncthreads();
  if (threadIdx.x == 0) {
    float s = 0.0f; int c = 0;
#pragma unroll
    for (int w = 0; w < kWavesPerBlock; ++w) { s += s_sum[w]; c += s_cnt[w]; }
    psum[blockIdx.x] = s;   // fully overwritten every launch -> deterministic,
    pcnt[blockIdx.x] = c;   // no dependence on stale workspace contents
  }
}

// Deterministic fixed-order reduction of the 4096 block partials, then the
// masked-mean: loss = sum(hard_loss) / max(1, n_hard).
__global__ __launch_bounds__(256) void triplet_final_kernel(
    const float* __restrict__ psum, const int* __restrict__ pcnt,
    float* __restrict__ out) {
  __shared__ float ss[256];
  __shared__ int   sc[256];
  float s = 0.0f; int c = 0;
  for (int i = threadIdx.x; i < kBlocks; i += 256) { s += psum[i]; c += pcnt[i]; }
  ss[threadIdx.x] = s;
  sc[threadIdx.x] = c;
  __syncthreads();
#pragma unroll
  for (int off = 128; off > 0; off >>= 1) {
    if ((int)threadIdx.x < off) {
      ss[threadIdx.x] += ss[threadIdx.x + off];
      sc[threadIdx.x] += sc[threadIdx.x + off];
    }
    __syncthreads();
  }
  if (threadIdx.x == 0) {
    const int n = sc[0] < 1 ? 1 : sc[0];
    out[0] = ss[0] / (float)n;
  }
}

extern "C" void kernel_launch(void* const* d_in, const int* in_sizes, int n_in,
                              void* d_out, int out_size, void* d_ws, size_t ws_size,
                              hipStream_t stream) {
  (void)in_sizes; (void)n_in; (void)out_size; (void)ws_size;
  const float* x    = (const float*)d_in[0];
  float*       psum = (float*)d_ws;                                   // 4096 f32
  int*         pcnt = (int*)((char*)d_ws + kBlocks * sizeof(float));  // 4096 i32

  triplet_partial_kernel<<<kBlocks, 256, 0, stream>>>(x, psum, pcnt);
  triplet_final_kernel<<<1, 256, 0, stream>>>(psum, pcnt, (float*)d_out);
}